// GNN_v2_18348100289075
// MI455X (gfx1250) — compile-verified
//
#include <hip/hip_runtime.h>
#include <math.h>

// ---------------------------------------------------------------------------
// GIN (3-layer) + pooling + prototype cosine loss for MI455X (gfx1250).
// FP32 end-to-end; GEMMs use V_WMMA_F32_16X16X4_F32 (wave32 WMMA).
// ---------------------------------------------------------------------------

#define D 128
constexpr int N_NODES  = 100000;
constexpr int N_EDGES  = 1600000;
constexpr int N_GRAPHS = 512;
constexpr float BN_EPS = 1e-5f;

typedef __attribute__((ext_vector_type(2))) float v2f;
typedef __attribute__((ext_vector_type(8))) float v8f;

// ---------------------------------------------------------------------------
__global__ void zero_f32(float* __restrict__ p, int n) {
  int i = blockIdx.x * blockDim.x + threadIdx.x;
  if (i < n) p[i] = 0.0f;
}

// ---------------------------------------------------------------------------
// Edge scatter-add: 32 lanes per edge, 4 consecutive features per lane.
// Wave reads one full 512B source row (coalesced), atomics to dest row.
__global__ void scatter_add(const float* __restrict__ h,
                            const int* __restrict__ src,
                            const int* __restrict__ dst,
                            float* agg) {
  long long tid = (long long)blockIdx.x * blockDim.x + threadIdx.x;
  int e = (int)(tid >> 5);
  if (e >= N_EDGES) return;
  int f = ((int)tid & 31) * 4;
  int s = src[e];
  int d = dst[e];
  const float4 v = *(const float4*)(h + (size_t)s * D + f);
  float* o = agg + (size_t)d * D + f;
  atomicAdd(o + 0, v.x);
  atomicAdd(o + 1, v.y);
  atomicAdd(o + 2, v.z);
  atomicAdd(o + 3, v.w);
}

// ---------------------------------------------------------------------------
// OUT[rows x 128] = relu( (A (+ A2 if FUSE)) @ W + bias )
//
// W row-major [K=128][N=128] is staged into LDS pre-paired along K:
//   ldsW2[(k/2)*128 + n] = { W[k][n], W[k+1][n] }   (k even)
// so each WMMA B operand is one aligned ds_load_b64 into an adjacent VGPR
// pair (no register shuffling in the hot loop).
//
// 8 waves/WG, each wave computes a 16x128 tile via 8 f32 accumulators and
// V_WMMA_F32_16X16X4_F32 (32 k-steps x 8 WMMAs).
//
// Lane mapping (ISA 7.12.2, f32):
//   A 16x4 : lane L (lrow=L&15, hf=L>>4): VGPR0 = A[M=lrow][K=k0+2*hf],
//                                         VGPR1 = A[M=lrow][K=k0+2*hf+1]
//   B 4x16 : VGPR0 = W[K=k0+2*hf][N=lrow], VGPR1 = W[K=k0+2*hf+1][N=lrow]
//   C/D    : VGPR r -> M = r + 8*hf, N = lrow
// NOTE: A2 and OUT may alias (wave-private rows; all loads precede stores),
// so they are intentionally NOT __restrict__.
template <bool FUSE>
__global__ __launch_bounds__(256) void gemm_relu(const float* __restrict__ A,
                                                 const float* A2,
                                                 const float* __restrict__ W,
                                                 const float* __restrict__ bias,
                                                 float* OUT, int n_rows) {
  __shared__ v2f ldsW2[(D / 2) * D];  // 64KB, K-paired layout
  {
    for (int i = threadIdx.x; i < (D / 2) * D; i += 256) {
      int kp = i >> 7;       // K pair index 0..63
      int n = i & (D - 1);   // 0..127
      v2f v;
      v.x = W[(size_t)(2 * kp) * D + n];
      v.y = W[(size_t)(2 * kp + 1) * D + n];
      ldsW2[i] = v;
    }
  }
  __syncthreads();

  int wave = threadIdx.x >> 5;
  int lane = threadIdx.x & 31;
  int rowbase = (blockIdx.x * 8 + wave) * 16;
  if (rowbase >= n_rows) return;  // wave-uniform; after barrier => safe

  int lrow = lane & 15;
  int hf = lane >> 4;

  const float* arow = A + (size_t)(rowbase + lrow) * D;
  const float* arow2 = FUSE ? (A2 + (size_t)(rowbase + lrow) * D) : nullptr;

  v8f acc[8];
#pragma unroll
  for (int j = 0; j < 8; j++) {
    float bv = bias[j * 16 + lrow];
#pragma unroll
    for (int r = 0; r < 8; r++) acc[j][r] = bv;
  }

  for (int k0 = 0; k0 < D; k0 += 4) {
    int ka = k0 + 2 * hf;  // even
    int kp = ka >> 1;      // pair index
    v2f a;
    a.x = arow[ka];
    a.y = arow[ka + 1];
    if (FUSE) {
      a.x += arow2[ka];
      a.y += arow2[ka + 1];
    }
    const v2f* brow = ldsW2 + (size_t)kp * D + lrow;
#pragma unroll
    for (int j = 0; j < 8; j++) {
      v2f b = brow[j * 16];
      acc[j] = __builtin_amdgcn_wmma_f32_16x16x4_f32(
          /*neg_a=*/false, a, /*neg_b=*/false, b,
          /*c_mod=*/(short)0, acc[j], /*reuse_a=*/false, /*reuse_b=*/false);
    }
  }

#pragma unroll
  for (int j = 0; j < 8; j++) {
#pragma unroll
    for (int r = 0; r < 8; r++) {
      int row = rowbase + r + 8 * hf;
      OUT[(size_t)row * D + j * 16 + lrow] = fmaxf(acc[j][r], 0.0f);
    }
  }
}

// ---------------------------------------------------------------------------
// BatchNorm statistics: per-column sum and sum-of-squares (block partials +
// atomics). 128 threads/block (one column each), 512 rows per block.
__global__ void bn_stats(const float* __restrict__ z, float* __restrict__ sum,
                         float* __restrict__ sumsq) {
  int c = threadIdx.x;  // 0..127
  int r0 = blockIdx.x * 512;
  int r1 = min(r0 + 512, N_NODES);
  float s = 0.0f, s2 = 0.0f;
  for (int r = r0; r < r1; ++r) {
    float v = z[(size_t)r * D + c];
    s += v;
    s2 += v * v;
  }
  atomicAdd(&sum[c], s);
  atomicAdd(&sumsq[c], s2);
}

__global__ void bn_apply(float* z, const float* __restrict__ sum,
                         const float* __restrict__ sumsq,
                         const float* __restrict__ gamma,
                         const float* __restrict__ beta) {
  size_t i = (size_t)blockIdx.x * blockDim.x + threadIdx.x;
  if (i >= (size_t)N_NODES * D) return;
  int c = (int)(i & (D - 1));
  const float invN = 1.0f / (float)N_NODES;
  float mu = sum[c] * invN;
  float var = sumsq[c] * invN - mu * mu;  // biased variance (torch BN)
  z[i] = (z[i] - mu) * rsqrtf(var + BN_EPS) * gamma[c] + beta[c];
}

// ---------------------------------------------------------------------------
// Per-graph pooling: mean-sum + count via f32 atomics, max via uint atomicMax
// on the float bit pattern (valid: post-ReLU values are >= 0).
__global__ void pool_kernel(const float* __restrict__ h,
                            const int* __restrict__ batch,
                            float* ssum, float* cnt, unsigned int* gmax) {
  long long tid = (long long)blockIdx.x * blockDim.x + threadIdx.x;
  int node = (int)(tid >> 5);
  if (node >= N_NODES) return;
  int f = ((int)tid & 31) * 4;
  int g = batch[node];
  const float4 v = *(const float4*)(h + (size_t)node * D + f);
  float* o = ssum + (size_t)g * D + f;
  atomicAdd(o + 0, v.x);
  atomicAdd(o + 1, v.y);
  atomicAdd(o + 2, v.z);
  atomicAdd(o + 3, v.w);
  unsigned int* m = gmax + (size_t)g * D + f;
  atomicMax(m + 0, __float_as_uint(v.x));
  atomicMax(m + 1, __float_as_uint(v.y));
  atomicMax(m + 2, __float_as_uint(v.z));
  atomicMax(m + 3, __float_as_uint(v.w));
  if (((int)tid & 31) == 0) atomicAdd(&cnt[g], 1.0f);
}

// ---------------------------------------------------------------------------
// Graph head: one 128-thread block per graph.
//   graph_x = [mean | max] (256) ; rep = graph_x @ Wg + bg ; normalize ;
//   cosine vs normalized prototypes ; squared loss terms -> atomicAdd(out).
__global__ void final_graph(const float* __restrict__ ssum,
                            const float* __restrict__ cnt,
                            const float* __restrict__ gmax,
                            const float* __restrict__ Wg,
                            const float* __restrict__ bg,
                            const float* __restrict__ pb,
                            const float* __restrict__ pm,
                            const int* __restrict__ y, float* out) {
  __shared__ float gx[2 * D];
  __shared__ float red[D];
  int g = blockIdx.x;
  int t = threadIdx.x;

  float c = cnt[g];
  float inv = 1.0f / fmaxf(c, 1.0f);
  gx[t] = ssum[(size_t)g * D + t] * inv;
  gx[D + t] = gmax[(size_t)g * D + t];
  __syncthreads();

  float acc = bg[t];
  for (int k = 0; k < 2 * D; k++) acc += gx[k] * Wg[(size_t)k * D + t];

  auto reduce = [&](float v) -> float {
    red[t] = v;
    __syncthreads();
    for (int s = 64; s > 0; s >>= 1) {
      if (t < s) red[t] += red[t + s];
      __syncthreads();
    }
    float r = red[0];
    __syncthreads();
    return r;
  };

  float n2 = reduce(acc * acc);
  float feat = acc / fmaxf(sqrtf(n2), 1e-12f);

  float pbv = pb[t];
  float pmv = pm[t];
  float nb = reduce(pbv * pbv);
  float nm = reduce(pmv * pmv);
  pbv /= fmaxf(sqrtf(nb), 1e-12f);
  pmv /= fmaxf(sqrtf(nm), 1e-12f);

  float cb = reduce(feat * pbv);
  float cm = reduce(feat * pmv);

  if (t == 0) {
    bool mal = (y[g] == 1);
    float tb = mal ? cb * cb : (1.0f - cb) * (1.0f - cb);
    float tm = mal ? (1.0f - cm) * (1.0f - cm) : cm * cm;
    atomicAdd(out, tb + tm);
  }
}

// ---------------------------------------------------------------------------
extern "C" void kernel_launch(void* const* d_in, const int* in_sizes, int n_in,
                              void* d_out, int out_size, void* d_ws,
                              size_t ws_size, hipStream_t stream) {
  const float* x  = (const float*)d_in[0];
  const int* ei   = (const int*)d_in[1];
  const int* batch = (const int*)d_in[2];
  const int* y    = (const int*)d_in[3];
  const float* W1 = (const float*)d_in[4];
  const float* b1 = (const float*)d_in[5];
  const float* W2 = (const float*)d_in[6];
  const float* b2 = (const float*)d_in[7];
  const float* gam = (const float*)d_in[8];
  const float* bet = (const float*)d_in[9];
  const float* Wg = (const float*)d_in[10];
  const float* bg = (const float*)d_in[11];
  const float* pb = (const float*)d_in[12];
  const float* pm = (const float*)d_in[13];
  const int* src = ei;            // edge_index[0]
  const int* dst = ei + N_EDGES;  // edge_index[1]

  // Workspace layout (floats):
  //   agg  [N*D]   (also reused as MLP hidden 't')
  //   z0   [N*D]
  //   z1   [N*D]
  //   stats[256]   (sum | sumsq)
  //   ssum [G*D], cnt [G], gmax [G*D]   (contiguous -> single zero pass)
  float* ws = (float*)d_ws;
  const size_t NF = (size_t)N_NODES * D;
  float* agg = ws;
  float* z0 = ws + NF;
  float* z1 = ws + 2 * NF;
  float* stats = ws + 3 * NF;
  float* ssum = stats + 256;
  float* cnt = ssum + (size_t)N_GRAPHS * D;
  float* gmax = cnt + N_GRAPHS;

  auto zero = [&](float* p, int n) {
    zero_f32<<<(n + 255) / 256, 256, 0, stream>>>(p, n);
  };

  const int gemm_blocks = (N_NODES / 16 + 7) / 8;          // 782
  const int scat_blocks = (int)(((long long)N_EDGES * 32) / 256);   // 200000
  const int pool_blocks = (int)(((long long)N_NODES * 32 + 255) / 256);

  const float* h = x;
  float* outs[3] = {z0, z1, z0};
  for (int l = 0; l < 3; l++) {
    zero(agg, (int)NF);
    scatter_add<<<scat_blocks, 256, 0, stream>>>(h, src, dst, agg);
    // t = relu((h + agg) @ W1 + b1); t aliases agg (safe: wave-private rows)
    gemm_relu<true><<<gemm_blocks, 256, 0, stream>>>(
        h, agg, W1 + (size_t)l * D * D, b1 + l * D, agg, N_NODES);
    // z = relu(t @ W2 + b2)
    gemm_relu<false><<<gemm_blocks, 256, 0, stream>>>(
        agg, agg, W2 + (size_t)l * D * D, b2 + l * D, outs[l], N_NODES);
    if (l < 2) {
      zero(stats, 256);
      bn_stats<<<(N_NODES + 511) / 512, 128, 0, stream>>>(outs[l], stats,
                                                          stats + 128);
      bn_apply<<<(int)((NF + 255) / 256), 256, 0, stream>>>(
          outs[l], stats, stats + 128, gam + l * D, bet + l * D);
    }
    h = outs[l];
  }

  // Pooling (zero ssum|cnt|gmax in one pass: all-zero bits == 0.0f)
  zero(ssum, N_GRAPHS * D + N_GRAPHS + N_GRAPHS * D);
  pool_kernel<<<pool_blocks, 256, 0, stream>>>(h, batch, ssum, cnt,
                                               (unsigned int*)gmax);

  zero((float*)d_out, 1);
  final_graph<<<N_GRAPHS, 128, 0, stream>>>(ssum, cnt, gmax, Wg, bg, pb, pm, y,
                                            (float*)d_out);
}